// FirstGNN_27805618274378
// MI455X (gfx1250) — compile-verified
//
#include <hip/hip_runtime.h>

#define N_NODES  50000
#define N_EDGES  800000
#define IN_CH    128
#define HID      64
#define OUT_CH   2
#define N_GRAPHS 512

typedef __attribute__((ext_vector_type(2))) float v2f;
typedef __attribute__((ext_vector_type(8))) float v8f;

// ---------------------------------------------------------------- utilities
__global__ void gnn_fill_kernel(float* __restrict__ p, float v, int n) {
    int i = blockIdx.x * blockDim.x + threadIdx.x;
    if (i < n) p[i] = v;
}

__global__ void gnn_deg_accum_kernel(const long long* __restrict__ dst,
                                     float* __restrict__ deg, int E) {
    int i = blockIdx.x * blockDim.x + threadIdx.x;
    if (i < E) atomicAdd(&deg[(int)dst[i]], 1.0f);
}

__global__ void gnn_rsqrt_kernel(float* __restrict__ p, int n) {
    int i = blockIdx.x * blockDim.x + threadIdx.x;
    if (i < n) p[i] = rsqrtf(p[i]);   // deg >= 1 always (self loops)
}

// ------------------------------------------------- fp32 WMMA GEMM, C = A*B
// A: [M,K] row-major, B: [K,64] row-major, C: [M,64]. M%16==0, K%4==0.
// B is staged into LDS in k-pair interleaved layout:
//   sB[(k>>1)*128 + col*2 + (k&1)]
// so each lane's B fragment {B[2p][col], B[2p+1][col]} is one ds_load_b64.
// One wave32 computes a 16x64 strip of C (4 accumulators): per k-step the
// loop is 1 global_load_b64 (A) + 4 ds_load_b64 (B) + 4 WMMAs.
__global__ void gnn_gemm_wmma_f32(const float* __restrict__ A,
                                  const float* __restrict__ B,
                                  float* __restrict__ C,
                                  int M, int K) {
    extern __shared__ float sB[];                 // K*64 floats, interleaved
    const int tid = threadIdx.x;

    // cooperative stage of the weight matrix into interleaved LDS layout
    const int nvec = (K * HID) >> 2;              // float4 elements of B
    const float4* __restrict__ Bv = (const float4*)B;
    for (int i = tid; i < nvec; i += blockDim.x) {
        int k  = i >> 4;                          // K row (16 float4 per row)
        int n0 = (i & 15) << 2;                   // starting column
        float4 v = Bv[i];
        float* d = sB + (k >> 1) * 128 + (k & 1);
        d[(n0 + 0) * 2] = v.x;
        d[(n0 + 1) * 2] = v.y;
        d[(n0 + 2) * 2] = v.z;
        d[(n0 + 3) * 2] = v.w;
    }
    __syncthreads();

    const int lane  = tid & 31;
    const int wave  = tid >> 5;
    const int strip = blockIdx.x * (blockDim.x >> 5) + wave;  // 16-row strip
    if (strip >= (M >> 4)) return;                // wave-uniform guard

    const int half = lane >> 4;                   // 0: lanes 0-15, 1: 16-31
    const int l15  = lane & 15;
    const float* __restrict__ arow = A + (size_t)(strip * 16 + l15) * K;

    v8f acc0 = {}, acc1 = {}, acc2 = {}, acc3 = {};
    for (int k0 = 0; k0 < K; k0 += 4) {
        const int p = (k0 >> 1) + half;           // k-pair index: ka = 2p
        v2f a = *(const v2f*)(arow + 2 * p);      // one b64: A[row][2p..2p+1]
        const float* bp = sB + p * 128 + l15 * 2; // interleaved pair base
        v2f b0 = *(const v2f*)(bp + 0);           // cols l15,    rows 2p,2p+1
        v2f b1 = *(const v2f*)(bp + 32);          // cols l15+16
        v2f b2 = *(const v2f*)(bp + 64);          // cols l15+32
        v2f b3 = *(const v2f*)(bp + 96);          // cols l15+48
        acc0 = __builtin_amdgcn_wmma_f32_16x16x4_f32(false, a, false, b0, (short)0, acc0, false, false);
        acc1 = __builtin_amdgcn_wmma_f32_16x16x4_f32(false, a, false, b1, (short)0, acc1, false, false);
        acc2 = __builtin_amdgcn_wmma_f32_16x16x4_f32(false, a, false, b2, (short)0, acc2, false, false);
        acc3 = __builtin_amdgcn_wmma_f32_16x16x4_f32(false, a, false, b3, (short)0, acc3, false, false);
    }

    // C/D layout: VGPR i -> row = strip*16 + i + 8*half, col = tile*16 + l15
    float* __restrict__ crow = C + (size_t)(strip * 16 + half * 8) * HID + l15;
    #pragma unroll
    for (int i = 0; i < 8; ++i) {
        crow[i * HID + 0]  = acc0[i];
        crow[i * HID + 16] = acc1[i];
        crow[i * HID + 32] = acc2[i];
        crow[i * HID + 48] = acc3[i];
    }
}

// ----------------------------------- edge scatter: out[dst] += h[src]*norm
// 16 threads per edge, 4 channels each: coalesced float4 gather + 4 atomics.
__global__ void gnn_scatter_kernel(const float* __restrict__ h,
                                   const long long* __restrict__ src,
                                   const long long* __restrict__ dst,
                                   const float* __restrict__ dinv,
                                   float* __restrict__ out, int E) {
    int t  = blockIdx.x * blockDim.x + threadIdx.x;
    int e  = t >> 4;
    int cg = (t & 15) << 2;
    if (e >= E) return;
    int s = (int)src[e], d = (int)dst[e];
    float norm = dinv[s] * dinv[d];
    float4 hv = *(const float4*)(h + (size_t)s * HID + cg);
    float* op = out + (size_t)d * HID + cg;
    atomicAdd(op + 0, hv.x * norm);
    atomicAdd(op + 1, hv.y * norm);
    atomicAdd(op + 2, hv.z * norm);
    atomicAdd(op + 3, hv.w * norm);
}

// ------------------- self-loop term + bias + ReLU: out = relu(out + h*dinv^2 + b)
__global__ void gnn_finalize_kernel(const float* __restrict__ h,
                                    const float* __restrict__ dinv,
                                    const float* __restrict__ bias,
                                    float* __restrict__ out, int n) {
    int i = blockIdx.x * blockDim.x + threadIdx.x;
    if (i >= n) return;
    int node = i >> 6;            // HID == 64
    int c    = i & 63;
    float di = dinv[node];
    float v  = out[i] + h[i] * di * di + bias[c];
    out[i] = fmaxf(v, 0.0f);
}

// ------------------------------------------------------- global mean pool
__global__ void gnn_pool_kernel(const float* __restrict__ h,
                                const long long* __restrict__ batch,
                                float* __restrict__ pool,
                                float* __restrict__ cnt, int n) {
    int i = blockIdx.x * blockDim.x + threadIdx.x;
    if (i >= n) return;
    int node = i >> 6;
    int c    = i & 63;
    int g = (int)batch[node];
    atomicAdd(&pool[g * HID + c], h[i]);
    if (c == 0) atomicAdd(&cnt[g], 1.0f);
}

// ------------------------------------------------------------ linear head
__global__ void gnn_head_kernel(const float* __restrict__ pool,
                                const float* __restrict__ cnt,
                                const float* __restrict__ Wl,
                                const float* __restrict__ bl,
                                float* __restrict__ out) {
    int i = blockIdx.x * blockDim.x + threadIdx.x;
    if (i >= N_GRAPHS * OUT_CH) return;
    int g = i / OUT_CH, o = i % OUT_CH;
    float inv = 1.0f / fmaxf(cnt[g], 1.0f);
    float s = bl[o];
    #pragma unroll 8
    for (int c = 0; c < HID; ++c)
        s += pool[g * HID + c] * inv * Wl[c * OUT_CH + o];
    out[i] = s;
}

extern "C" void kernel_launch(void* const* d_in, const int* in_sizes, int n_in,
                              void* d_out, int out_size, void* d_ws, size_t ws_size,
                              hipStream_t stream) {
    const float*     x     = (const float*)d_in[0];
    const long long* ei    = (const long long*)d_in[1];   // int64 [2, E]
    const long long* batch = (const long long*)d_in[2];   // int64 [N]
    const float*     W1    = (const float*)d_in[3];
    const float*     b1    = (const float*)d_in[4];
    const float*     W2    = (const float*)d_in[5];
    const float*     b2    = (const float*)d_in[6];
    const float*     Wl    = (const float*)d_in[7];
    const float*     bl    = (const float*)d_in[8];
    float*           out   = (float*)d_out;

    const long long* srcp = ei;             // row 0
    const long long* dstp = ei + N_EDGES;   // row 1

    // workspace layout (floats): dinv | bufA | bufB | pool | cnt
    float* dinv = (float*)d_ws;
    float* bufA = dinv + 50176;                       // 16B-aligned offset
    float* bufB = bufA + (size_t)N_NODES * HID;
    float* pool = bufB + (size_t)N_NODES * HID;
    float* cnt  = pool + N_GRAPHS * HID;

    const int B = 256;
    const int nNH      = N_NODES * HID;               // 3.2M
    const int gNodes   = (N_NODES + B - 1) / B;
    const int gEdges   = (N_EDGES + B - 1) / B;
    const int gNH      = (nNH + B - 1) / B;
    const int gScatter = (N_EDGES * 16 + B - 1) / B;  // 16 threads / edge
    const int strips   = N_NODES / 16;                // 3125 16-row strips
    const int gGemm    = (strips + 7) / 8;            // 8 waves / block

    // degree -> dinv (deg includes self loop)
    gnn_fill_kernel<<<gNodes, B, 0, stream>>>(dinv, 1.0f, N_NODES);
    gnn_deg_accum_kernel<<<gEdges, B, 0, stream>>>(dstp, dinv, N_EDGES);
    gnn_rsqrt_kernel<<<gNodes, B, 0, stream>>>(dinv, N_NODES);

    // ---- layer 1: h1 = relu(scatter(x@W1) + selfloop + b1)
    gnn_gemm_wmma_f32<<<gGemm, B, IN_CH * HID * sizeof(float), stream>>>(
        x, W1, bufA, N_NODES, IN_CH);
    gnn_fill_kernel<<<gNH, B, 0, stream>>>(bufB, 0.0f, nNH);
    gnn_scatter_kernel<<<gScatter, B, 0, stream>>>(bufA, srcp, dstp, dinv, bufB, N_EDGES);
    gnn_finalize_kernel<<<gNH, B, 0, stream>>>(bufA, dinv, b1, bufB, nNH);

    // ---- layer 2: h2 = relu(scatter(h1@W2) + selfloop + b2)
    gnn_gemm_wmma_f32<<<gGemm, B, HID * HID * sizeof(float), stream>>>(
        bufB, W2, bufA, N_NODES, HID);
    gnn_fill_kernel<<<gNH, B, 0, stream>>>(bufB, 0.0f, nNH);
    gnn_scatter_kernel<<<gScatter, B, 0, stream>>>(bufA, srcp, dstp, dinv, bufB, N_EDGES);
    gnn_finalize_kernel<<<gNH, B, 0, stream>>>(bufA, dinv, b2, bufB, nNH);

    // ---- mean pool + head
    gnn_fill_kernel<<<(N_GRAPHS * HID + N_GRAPHS + B - 1) / B, B, 0, stream>>>(
        pool, 0.0f, N_GRAPHS * HID + N_GRAPHS);       // zeros pool and cnt together
    gnn_pool_kernel<<<gNH, B, 0, stream>>>(bufB, batch, pool, cnt, nNH);
    gnn_head_kernel<<<(N_GRAPHS * OUT_CH + B - 1) / B, B, 0, stream>>>(
        pool, cnt, Wl, bl, out);
}